// HarmonicDDSPEngine_43138651521400
// MI455X (gfx1250) — compile-verified
//
#include <hip/hip_runtime.h>
#include <math.h>

typedef __attribute__((ext_vector_type(2))) float v2f;
typedef __attribute__((ext_vector_type(8))) float v8f;

#define SRATE     44100.0f
#define T_SAMPLES 176400
#define NHARM     64
#define NBATCH    16
#define NBANDS    65
#define QMOD      176399u          // (T-1); phase(k,t) turns = (1760*k*t)/176399
#define PMUL      1760u            // F0 * (T/SR) = 440*4
#define PSTRIDE   16               // params stride (floats) per batch

// params per batch (stride 16 floats):
// 0:t1(=a) 1:t2(=a+d) 2:t3(=a+d+s) 3:t4(=a+d+s+r)
// 4:invA   5:slopeD(=(sus-1)/max(d-1,1)) 6:sus 7:srel(=sus/max(r-1,1))
// 8:nlevel 9:gain
__global__ void ddsp_setup(const float* __restrict__ noise_bands, // (16,65)
                           const float* __restrict__ adsr,        // (16,4)
                           const float* __restrict__ gain,        // (16,1)
                           float* __restrict__ params,            // (16,16)
                           float* __restrict__ bmax)              // (16)
{
    int b = threadIdx.x;
    if (b < NBATCH) {
        float attack  = adsr[b * 4 + 0];
        float decay   = adsr[b * 4 + 1];
        float sus     = adsr[b * 4 + 2];
        float release = adsr[b * 4 + 3];
        int a = (int)floorf(attack  * 0.5f * SRATE) + 1;
        int d = (int)floorf(decay   * 0.5f * SRATE) + 1;
        int r = (int)floorf(release * 0.5f * SRATE) + 1;
        int total = a + d + r;
        if (total > T_SAMPLES) {
            float scale = (float)T_SAMPLES / (float)total;
            a = (int)floorf((float)a * scale);
            d = (int)floorf((float)d * scale);
            r = (int)floorf((float)r * scale);
        }
        int s = T_SAMPLES - (a + d + r);
        if (s < 0) s = 0;

        float af = (float)a, df = (float)d, rf = (float)r, sf = (float)s;
        float nl = 0.0f;
        for (int i = 0; i < NBANDS; ++i) nl += noise_bands[b * NBANDS + i];
        nl = (nl / (float)NBANDS) * 0.1f;

        params[b * PSTRIDE + 0] = af;
        params[b * PSTRIDE + 1] = af + df;
        params[b * PSTRIDE + 2] = af + df + sf;
        params[b * PSTRIDE + 3] = af + df + sf + rf;
        params[b * PSTRIDE + 4] = (af > 1.0f) ? 1.0f / fmaxf(af - 1.0f, 1.0f) : 0.0f;
        params[b * PSTRIDE + 5] = (sus - 1.0f) / fmaxf(df - 1.0f, 1.0f);
        params[b * PSTRIDE + 6] = sus;
        params[b * PSTRIDE + 7] = sus / fmaxf(rf - 1.0f, 1.0f);
        params[b * PSTRIDE + 8] = nl;
        params[b * PSTRIDE + 9] = gain[b];
        bmax[b] = 0.0f;
    }
}

// ---------------------------------------------------------------------------
// Main: per-wave 16x16 tile (16 batches x 16 samples).
// D[b,t] = sum_k hdist[b,k]*sin(2*pi*(1760*(k+1)*t)/176399) via 16 f32 WMMAs.
// Phase kept as an exact integer residue P=(1760*t) mod Q, advanced
// incrementally (2 int ops/tile); each sin = v_mul + v_fract + v_sin.
// ---------------------------------------------------------------------------
__global__ void __launch_bounds__(256) ddsp_main(
    const float* __restrict__ hdist,   // (16,64)
    const float* __restrict__ noise,   // (16,T)
    const float* __restrict__ params,  // (16,16)
    float* __restrict__ out,           // (16,T)
    float* __restrict__ bmax)          // (16), float bits via atomicMax(int)
{
    const int lane  = threadIdx.x & 31;
    const int wib   = threadIdx.x >> 5;
    const int wpb   = blockDim.x >> 5;
    const int gwave = blockIdx.x * wpb + wib;
    const int nwave = gridDim.x * wpb;

    const int n    = lane & 15;  // N (time-in-tile) for B/C; M (batch) for A
    const int half = lane >> 4;  // selects K pair (A/B) / batch group (C)

    // A-matrix (16x4 f32) preload, 16 K-steps. ISA layout:
    // lanes 0-15: M=lane, K=base+0 (.x), base+1 (.y); lanes 16-31: K=base+2/+3.
    v2f A[16];
    {
        const int kofs = half * 2;
        #pragma unroll
        for (int s = 0; s < 16; ++s) {
            const int kb = s * 4 + kofs;
            A[s].x = hdist[n * NHARM + kb];
            A[s].y = hdist[n * NHARM + kb + 1];
        }
    }

    // Per-lane epilogue params for the 8 batches this lane owns in C
    // (C layout: VGPR r holds M = r + 8*half, N = lane&15).
    float pt1[8], pt2[8], pt3[8], pt4[8], pinvA[8], psD[8], psus[8], psrel[8],
          pnl[8], pg[8];
    #pragma unroll
    for (int rr = 0; rr < 8; ++rr) {
        const int m = rr + 8 * half;
        pt1[rr]  = params[m * PSTRIDE + 0];
        pt2[rr]  = params[m * PSTRIDE + 1];
        pt3[rr]  = params[m * PSTRIDE + 2];
        pt4[rr]  = params[m * PSTRIDE + 3];
        pinvA[rr]= params[m * PSTRIDE + 4];
        psD[rr]  = params[m * PSTRIDE + 5];
        psus[rr] = params[m * PSTRIDE + 6];
        psrel[rr]= params[m * PSTRIDE + 7];
        pnl[rr]  = params[m * PSTRIDE + 8];
        pg[rr]   = params[m * PSTRIDE + 9];
    }

    float lmax[8];
    #pragma unroll
    for (int rr = 0; rr < 8; ++rr) lmax[rr] = 0.0f;

    // Exact integer phase residue for this lane's time sample.
    int t = gwave * 16 + n;
    unsigned int P = (PMUL * (unsigned int)t) % QMOD;               // < 2^31, exact
    const unsigned int D =
        (unsigned int)((unsigned long long)PMUL * 16ull *
                       (unsigned long long)nwave % (unsigned long long)QMOD);
    const float invQ = 1.0f / (float)QMOD;
    const float hf   = (float)(2 * half);

    const int ntiles = T_SAMPLES / 16; // 11025, exact
    for (int tile = gwave; tile < ntiles; tile += nwave) {
        const float f1 = (float)P * invQ;   // turns per unit harmonic, in [0,1)

        v8f C = {};
        #pragma unroll
        for (int s = 0; s < 16; ++s) {
            // B (4x16 f32): VGPR0 lanes0-15 K=base+0, lanes16-31 K=base+2;
            //               VGPR1 K=base+1 / base+3.  Harmonic = Kidx+1.
            const float ke = (float)(4 * s + 1) + hf;
            v2f B;
            // v_sin_f32 computes sin(2*pi*x): feed fractional turns directly.
            B.x = __builtin_amdgcn_sinf(__builtin_amdgcn_fractf(ke * f1));
            B.y = __builtin_amdgcn_sinf(__builtin_amdgcn_fractf((ke + 1.0f) * f1));
            C = __builtin_amdgcn_wmma_f32_16x16x4_f32(
                    /*neg_a=*/false, A[s], /*neg_b=*/false, B,
                    /*c_mod=*/(short)0, C, /*reuse_a=*/false, /*reuse_b=*/false);
        }

        // Fused epilogue: noise + ADSR env (FMA/select only) + gain.
        const float fi = (float)t;
        #pragma unroll
        for (int rr = 0; rr < 8; ++rr) {
            const int m   = rr + 8 * half;
            const float nv  = fmaf(noise[m * T_SAMPLES + t], 2.0f, -1.0f) * pnl[rr];
            const float att = fi * pinvA[rr];
            const float dec = fmaf(fi - pt1[rr], psD[rr], 1.0f);
            const float rel = fmaf(pt3[rr] - fi, psrel[rr], psus[rr]);
            float env = (fi < pt4[rr]) ? rel : 0.0f;
            env = (fi < pt3[rr]) ? psus[rr] : env;
            env = (fi < pt2[rr]) ? dec : env;
            env = (fi < pt1[rr]) ? att : env;
            const float sig = (C[rr] + nv) * env * pg[rr];
            out[m * T_SAMPLES + t] = sig;
            lmax[rr] = fmaxf(lmax[rr], fabsf(sig));
        }

        // Advance exact phase residue and time.
        t += nwave * 16;
        unsigned int Pn = P + D;
        P = (Pn >= QMOD) ? Pn - QMOD : Pn;
    }

    // Per-batch max across the 16 lanes sharing the same batch group.
    #pragma unroll
    for (int off = 8; off >= 1; off >>= 1) {
        #pragma unroll
        for (int rr = 0; rr < 8; ++rr)
            lmax[rr] = fmaxf(lmax[rr], __shfl_xor(lmax[rr], off, 32));
    }
    if (n == 0) {
        #pragma unroll
        for (int rr = 0; rr < 8; ++rr) {
            const int m = rr + 8 * half;
            atomicMax((int*)bmax + m, __float_as_int(lmax[rr])); // vals >= 0
        }
    }
}

// ---------------------------------------------------------------------------
// Normalize: out /= (max|out| per batch + 1e-5)
// ---------------------------------------------------------------------------
__global__ void ddsp_norm(float* __restrict__ out, const float* __restrict__ bmax)
{
    const int idx = blockIdx.x * blockDim.x + threadIdx.x;
    if (idx < NBATCH * T_SAMPLES) {
        const int b = idx / T_SAMPLES;
        out[idx] = out[idx] / (bmax[b] + 1e-5f);
    }
}

extern "C" void kernel_launch(void* const* d_in, const int* in_sizes, int n_in,
                              void* d_out, int out_size, void* d_ws, size_t ws_size,
                              hipStream_t stream)
{
    // setup_inputs order: base_audio(unused), harmonic_dist, noise_bands, adsr, gain, noise
    const float* hdist  = (const float*)d_in[1];
    const float* nbands = (const float*)d_in[2];
    const float* adsr   = (const float*)d_in[3];
    const float* gain   = (const float*)d_in[4];
    const float* noise  = (const float*)d_in[5];
    float* out = (float*)d_out;

    float* params = (float*)d_ws;                 // 16*16 floats
    float* bmax   = params + NBATCH * PSTRIDE;    // 16 floats

    ddsp_setup<<<1, 32, 0, stream>>>(nbands, adsr, gain, params, bmax);

    // 512 blocks x 8 waves = 4096 waves over 11025 tiles (grid-stride, full waves)
    ddsp_main<<<512, 256, 0, stream>>>(hdist, noise, params, out, bmax);

    const int nelem = NBATCH * T_SAMPLES;
    ddsp_norm<<<(nelem + 255) / 256, 256, 0, stream>>>(out, bmax);
}